// SNNLayer_80152679678494
// MI455X (gfx1250) — compile-verified
//
#include <hip/hip_runtime.h>

// CDNA5 / gfx1250: wave32, WMMA 16x16x4 f32 path.
// One wave == one grid cell (256x256 = 65536 cells). Memory-bound streaming
// kernel: every byte of input_matrix / layer_weights is touched once ->
// non-temporal loads + prefetch, fp32 WMMA for both batched matvec stages.

typedef __attribute__((ext_vector_type(2))) float v2f;
typedef __attribute__((ext_vector_type(8))) float v8f;

#define DECAY_FACTOR 0.8f
#define NN 32            // N (state size)
#define CELL_MAT (NN*NN) // 1024 floats per per-cell matrix

__device__ __forceinline__ v8f wmma_f32_16x16x4(v2f a, v2f b, v8f c) {
  // (neg_a, A, neg_b, B, c_mod, C, reuse_a, reuse_b)
  return __builtin_amdgcn_wmma_f32_16x16x4_f32(false, a, false, b,
                                               (short)0, c, false, false);
}

__global__ __launch_bounds__(256)
void snn_layer_wmma_kernel(const float* __restrict__ x,
                           const float* __restrict__ input_matrix,
                           const float* __restrict__ layer_weights,
                           float* __restrict__ out,
                           int n_cells) {
  const int lane = threadIdx.x & 31;
  const int wave = threadIdx.x >> 5;
  const int cell = blockIdx.x * 8 + wave;   // 8 waves / block, 1 cell / wave

  const int nl       = lane & 15;           // lane within half
  const int half     = lane >> 4;           // 0: lanes 0-15, 1: lanes 16-31
  const int krow_off = half * 2;            // K row pair handled by this half
  const bool row0    = (nl == 0);           // lanes holding A-matrix row M=0

  const float* M = input_matrix + (size_t)cell * CELL_MAT;

  // x vector for this cell: lane i holds x[i] (coalesced 128B load)
  const float xv = x[(size_t)cell * NN + lane];

  // Prefetch direction-0 weight matrix (4 KB: 32 lanes x 128 B cachelines)
  __builtin_prefetch(layer_weights + (size_t)cell * CELL_MAT + lane * 32, 0, 0);

  // ---------------- Stage 1: active = x @ M  (1x32 @ 32x32) ----------------
  v8f acc0 = {}; // N columns 0..15  (row 0 = active[0:16])
  v8f acc1 = {}; // N columns 16..31
  #pragma unroll
  for (int kb = 0; kb < NN; kb += 4) {
    // A (16x4 f32): only row M=0 populated with x[kb..kb+3]
    float ax = __shfl(xv, kb + krow_off,     32);
    float ay = __shfl(xv, kb + krow_off + 1, 32);
    v2f a;
    a.x = row0 ? ax : 0.0f;
    a.y = row0 ? ay : 0.0f;

    // B (4x16 f32), N half 0: v0 = rows K=kb+0 / kb+2, v1 = rows K=kb+1 / kb+3
    const float* p0 = M + (kb + krow_off) * NN + nl;
    v2f b0;
    b0.x = __builtin_nontemporal_load(p0);
    b0.y = __builtin_nontemporal_load(p0 + NN);
    acc0 = wmma_f32_16x16x4(a, b0, acc0);

    // B, N half 1 (columns 16..31)
    const float* p1 = p0 + 16;
    v2f b1;
    b1.x = __builtin_nontemporal_load(p1);
    b1.y = __builtin_nontemporal_load(p1 + NN);
    acc1 = wmma_f32_16x16x4(a, b1, acc1);
  }

  // Gather active[] across the wave: D row 0 lives in component 0, lanes 0-15.
  // active[lane] : lane<16 -> acc0[0]@lane ; lane>=16 -> acc1[0]@(lane-16)
  const float act = (lane < 16) ? acc0[0] : __shfl(acc1[0], nl, 32);

  // -------- Stage 2: out = 0.8 * sum_d active @ W_d  (4 chained GEMVs) -----
  v8f o0 = {};
  v8f o1 = {};
  #pragma unroll
  for (int d = 0; d < 4; ++d) {
    const float* Wd = layer_weights + ((size_t)d * n_cells + cell) * CELL_MAT;
    if (d < 3) {
      // prefetch next direction's 4 KB matrix while this one is consumed
      __builtin_prefetch(
          layer_weights + ((size_t)(d + 1) * n_cells + cell) * CELL_MAT + lane * 32,
          0, 0);
    }
    #pragma unroll
    for (int kb = 0; kb < NN; kb += 4) {
      float ax = __shfl(act, kb + krow_off,     32);
      float ay = __shfl(act, kb + krow_off + 1, 32);
      v2f a;
      a.x = row0 ? ax : 0.0f;
      a.y = row0 ? ay : 0.0f;

      const float* p0 = Wd + (kb + krow_off) * NN + nl;
      v2f b0;
      b0.x = __builtin_nontemporal_load(p0);
      b0.y = __builtin_nontemporal_load(p0 + NN);
      o0 = wmma_f32_16x16x4(a, b0, o0);

      const float* p1 = p0 + 16;
      v2f b1;
      b1.x = __builtin_nontemporal_load(p1);
      b1.y = __builtin_nontemporal_load(p1 + NN);
      o1 = wmma_f32_16x16x4(a, b1, o1);
    }
  }

  float r = (lane < 16) ? o0[0] : __shfl(o1[0], nl, 32);
  r *= DECAY_FACTOR;

  // streamed one-touch output: non-temporal store, coalesced 128B per wave
  __builtin_nontemporal_store(r, out + (size_t)cell * NN + lane);
}

extern "C" void kernel_launch(void* const* d_in, const int* in_sizes, int n_in,
                              void* d_out, int out_size, void* d_ws, size_t ws_size,
                              hipStream_t stream) {
  (void)n_in; (void)out_size; (void)d_ws; (void)ws_size;

  const float* x  = (const float*)d_in[0];             // (L, W, 1, N)
  const float* im = (const float*)d_in[1];             // (L, W, N, N)
  const float* lw = (const float*)d_in[2];             // (4, L, W, N, N)
  float* out = (float*)d_out;                          // (L, W, 1, N)

  const int n_cells = in_sizes[0] / NN;                // L*W = 65536
  const int blocks  = n_cells / 8;                     // 8 waves (cells) / block

  snn_layer_wmma_kernel<<<dim3(blocks), dim3(256), 0, stream>>>(
      x, im, lw, out, n_cells);
}